// GATPathogenicityModel_87428354277655
// MI455X (gfx1250) — compile-verified
//
#include <hip/hip_runtime.h>
#include <hip/hip_bf16.h>
#include <math.h>

typedef __attribute__((ext_vector_type(2))) float v2f;
typedef __attribute__((ext_vector_type(8))) float v8f;

#define HEADS 4
#define HID   32
#define GRP   64

// ---------------------------------------------------------------------------
// WMMA f32 GEMM with LDS-staged weights:
//   C[n,M] = A[n,K] @ B[K,M],  M = MT*16, K multiple of 4, n multiple of 16.
// Per block (256 thr / 8 waves): async-copy all of B into LDS
// (global_load_async_to_lds_b128, ASYNCcnt), then each wave computes one
// 16-row strip across all MT column tiles => A is read exactly once.
// ---------------------------------------------------------------------------
template<int K, int MT>
__global__ __launch_bounds__(256)
void gemm_wmma_lds(const float* __restrict__ A, const float* __restrict__ B,
                   float* __restrict__ C, int n) {
    constexpr int M = MT * 16;
    __shared__ float sW[K * M];                    // 64KB (MT=8) / 16KB (MT=2)

    // --- cooperative async load of W panel into LDS (all threads, pre-guard)
    unsigned lds_base = (unsigned)(uintptr_t)(&sW[0]);
    for (int i = threadIdx.x; i < (K * M) / 4; i += 256) {
        unsigned loff = lds_base + (unsigned)i * 16u;
        const float* g = B + (size_t)i * 4;
        asm volatile("global_load_async_to_lds_b128 %0, %1, off"
                     :: "v"(loff), "v"(g) : "memory");
    }
    asm volatile("s_wait_asynccnt 0x0" ::: "memory");
    __syncthreads();

    int wave  = threadIdx.x >> 5;
    int lane  = threadIdx.x & 31;
    int strip = blockIdx.x * 8 + wave;             // one 16-row strip per wave
    if (strip >= (n >> 4)) return;                 // wave-uniform: EXEC all-1s
    int row0 = strip << 4;
    int hf   = lane >> 4;                          // 0: K pair {0,1}, 1: {2,3}
    int lr   = lane & 15;

    v8f acc[MT];
#pragma unroll
    for (int t = 0; t < MT; ++t) acc[t] = 0.0f;

    const float* ap = A + (size_t)(row0 + lr) * K + 2 * hf;   // A[row][k0+2h+{0,1}]
    const float* bb = &sW[(2 * hf) * M + lr];                 // B[k0+2h+{0,1}][col]
    for (int k0 = 0; k0 < K; k0 += 4) {
        v2f a;
        a.x = ap[0];  a.y = ap[1];
        ap += 4;
        const float* bp = bb + (size_t)k0 * M;
#pragma unroll
        for (int t = 0; t < MT; ++t) {
            v2f b;
            b.x = bp[t * 16];                      // ds_load from LDS
            b.y = bp[t * 16 + M];
            acc[t] = __builtin_amdgcn_wmma_f32_16x16x4_f32(false, a, false, b,
                                                           (short)0, acc[t],
                                                           false, false);
        }
    }
    // D layout: VGPR i -> row0 + i + 8*hf, col = col0 + lr
#pragma unroll
    for (int t = 0; t < MT; ++t) {
        float* cp = C + (size_t)(row0 + 8 * hf) * M + t * 16 + lr;
#pragma unroll
        for (int i = 0; i < 8; ++i) cp[(size_t)i * M] = acc[t][i];
    }
}

// ---------------------------------------------------------------------------
// Attention scores: als/ald[node,h] = sum_c feat[node,h,c] * a_{s,d}[h,c]
// ---------------------------------------------------------------------------
__global__ void score_kernel(const float* __restrict__ feat, const float* __restrict__ a_s,
                             const float* __restrict__ a_d, float* __restrict__ als,
                             float* __restrict__ ald, int n, int H, int C) {
    int idx = blockIdx.x * blockDim.x + threadIdx.x;
    if (idx >= n * H) return;
    int node = idx / H, h = idx % H;
    const float* f  = feat + (size_t)node * H * C + (size_t)h * C;
    const float* w1 = a_s + (size_t)h * C;
    const float* w2 = a_d + (size_t)h * C;
    float s1 = 0.f, s2 = 0.f;
    for (int j = 0; j < C; ++j) { float v = f[j]; s1 += v * w1[j]; s2 += v * w2[j]; }
    als[idx] = s1;
    ald[idx] = s2;
}

__global__ void fill_kernel(float* __restrict__ p, float v, int n) {
    int i = blockIdx.x * blockDim.x + threadIdx.x;
    if (i < n) p[i] = v;
}

__device__ __forceinline__ void atomicMaxFloat(float* addr, float val) {
    if (val >= 0.f) atomicMax((int*)addr, __float_as_int(val));
    else            atomicMin((unsigned int*)addr, __float_as_uint(val));
}

__device__ __forceinline__ float leaky(float v) { return v > 0.f ? v : 0.2f * v; }

// Pass A: segment max of leaky_relu(al_s[src]+al_d[dst]) over dst
__global__ void edge_max_kernel(const int* __restrict__ src, const int* __restrict__ dst,
                                const float* __restrict__ als, const float* __restrict__ ald,
                                float* __restrict__ m, int E_, int H) {
    long long idx = (long long)blockIdx.x * blockDim.x + threadIdx.x;
    if (idx >= (long long)E_ * H) return;
    int e = (int)(idx / H), h = (int)(idx % H);
    int s = src[e], d = dst[e];
    float v = leaky(als[(size_t)s * H + h] + ald[(size_t)d * H + h]);
    atomicMaxFloat(&m[(size_t)d * H + h], v);
}

// Pass B: ex = exp(e - m[dst]); segment-sum into s
__global__ void edge_expsum_kernel(const int* __restrict__ src, const int* __restrict__ dst,
                                   const float* __restrict__ als, const float* __restrict__ ald,
                                   const float* __restrict__ m, float* __restrict__ ssum,
                                   float* __restrict__ ex, int E_, int H) {
    long long idx = (long long)blockIdx.x * blockDim.x + threadIdx.x;
    if (idx >= (long long)E_ * H) return;
    int e = (int)(idx / H), h = (int)(idx % H);
    int s = src[e], d = dst[e];
    float v = leaky(als[(size_t)s * H + h] + ald[(size_t)d * H + h]);
    float exv = __expf(v - m[(size_t)d * H + h]);
    ex[idx] = exv;
    atomicAdd(&ssum[(size_t)d * H + h], exv);
}

// Pass C: agg[dst, j..j+3] += feat[src, j..j+3] * alpha(edge, head(j))
// One thread per (edge, 4 channels); channels stay within one head (C=32).
__global__ void edge_agg_kernel(const int* __restrict__ src, const int* __restrict__ dst,
                                const float* __restrict__ feat, const float* __restrict__ ex,
                                const float* __restrict__ ssum, float* __restrict__ agg,
                                int E_, int H, int HC) {
    int Q = HC >> 2;
    long long idx = (long long)blockIdx.x * blockDim.x + threadIdx.x;
    if (idx >= (long long)E_ * Q) return;
    int e = (int)(idx / Q);
    int j = ((int)(idx % Q)) << 2;
    int h = j / (HC / H);
    int s = src[e], d = dst[e];
    float alpha = ex[(size_t)e * H + h] / (ssum[(size_t)d * H + h] + 1e-16f);
    const float4 f = *(const float4*)(feat + (size_t)s * HC + j);
    float* ap = agg + (size_t)d * HC + j;
    atomicAdd(ap + 0, f.x * alpha);
    atomicAdd(ap + 1, f.y * alpha);
    atomicAdd(ap + 2, f.z * alpha);
    atomicAdd(ap + 3, f.w * alpha);
}

// Fused bias + LayerNorm + ELU, one wave32 per node (F = 32 or 128)
__global__ void ln_elu_kernel(const float* __restrict__ in, const float* __restrict__ bias,
                              const float* __restrict__ gain, const float* __restrict__ beta,
                              float* __restrict__ out, int n, int F) {
    int wave = (int)((blockIdx.x * blockDim.x + threadIdx.x) >> 5);
    int lane = threadIdx.x & 31;
    if (wave >= n) return;
    const float* row = in + (size_t)wave * F;
    int per = F >> 5;                               // elements per lane (1 or 4)
    float vals[4];
    float sum = 0.f, sq = 0.f;
    for (int i = 0; i < per; ++i) {
        int j = lane * per + i;
        float v = row[j] + bias[j];
        vals[i] = v; sum += v; sq += v * v;
    }
    for (int off = 16; off > 0; off >>= 1) {
        sum += __shfl_xor(sum, off, 32);
        sq  += __shfl_xor(sq,  off, 32);
    }
    float inv = 1.0f / (float)F;
    float mu  = sum * inv;
    float var = sq * inv - mu * mu;
    float rs  = rsqrtf(var + 1e-5f);
    for (int i = 0; i < per; ++i) {
        int j = lane * per + i;
        float y = (vals[i] - mu) * rs * gain[j] + beta[j];
        out[(size_t)wave * F + j] = y > 0.f ? y : expm1f(y);
    }
}

// Global mean pool (sum + count, divide in classifier)
__global__ void pool_kernel(const float* __restrict__ h, const int* __restrict__ batch,
                            float* __restrict__ pooled, float* __restrict__ cnt, int n) {
    int idx = blockIdx.x * blockDim.x + threadIdx.x;
    if (idx >= n * HID) return;
    int node = idx >> 5, c = idx & 31;
    int b = batch[node];
    atomicAdd(&pooled[(size_t)b * HID + c], h[(size_t)node * HID + c]);
    if (c == 0) atomicAdd(&cnt[b], 1.0f);
}

// Tiny classifier: hg -> relu(fc1) -> fc2, one thread per graph
__global__ void classify_kernel(const float* __restrict__ pooled, const float* __restrict__ cnt,
                                const float* __restrict__ w1, const float* __restrict__ b1,
                                const float* __restrict__ w2, const float* __restrict__ b2,
                                float* __restrict__ out) {
    int g = threadIdx.x;
    if (g >= GRP) return;
    float c = cnt[g]; c = c > 1.f ? c : 1.f;
    float hg[HID], z[HID / 2];
    for (int j = 0; j < HID; ++j) hg[j] = pooled[g * HID + j] / c;
    for (int k = 0; k < HID / 2; ++k) {
        float acc = b1[k];
        for (int j = 0; j < HID; ++j) acc += hg[j] * w1[j * (HID / 2) + k];
        z[k] = acc > 0.f ? acc : 0.f;
    }
    for (int q = 0; q < 2; ++q) {
        float acc = b2[q];
        for (int k = 0; k < HID / 2; ++k) acc += z[k] * w2[k * 2 + q];
        out[g * 2 + q] = acc;
    }
}

extern "C" void kernel_launch(void* const* d_in, const int* in_sizes, int n_in,
                              void* d_out, int out_size, void* d_ws, size_t ws_size,
                              hipStream_t stream) {
    const float* x      = (const float*)d_in[0];
    const int*   eidx   = (const int*)d_in[1];
    const int*   batch  = (const int*)d_in[2];
    const float* W0     = (const float*)d_in[3];
    const float* as0    = (const float*)d_in[4];
    const float* ad0    = (const float*)d_in[5];
    const float* b0     = (const float*)d_in[6];
    const float* ln0g   = (const float*)d_in[7];
    const float* ln0b   = (const float*)d_in[8];
    const float* W1     = (const float*)d_in[9];
    const float* as1    = (const float*)d_in[10];
    const float* ad1    = (const float*)d_in[11];
    const float* b1     = (const float*)d_in[12];
    const float* ln1g   = (const float*)d_in[13];
    const float* ln1b   = (const float*)d_in[14];
    const float* W2     = (const float*)d_in[15];
    const float* as2    = (const float*)d_in[16];
    const float* ad2    = (const float*)d_in[17];
    const float* b2     = (const float*)d_in[18];
    const float* ln2g   = (const float*)d_in[19];
    const float* ln2b   = (const float*)d_in[20];
    const float* fc1w   = (const float*)d_in[21];
    const float* fc1b   = (const float*)d_in[22];
    const float* fc2w   = (const float*)d_in[23];
    const float* fc2b   = (const float*)d_in[24];

    const int F_IN = 128, HC = HEADS * HID;          // 128
    const int N = in_sizes[0] / F_IN;                // 40000
    const int E = in_sizes[1] / 2;                   // 640000
    const int* src = eidx;
    const int* dst = eidx + E;

    // Workspace carve (floats)
    float* ws   = (float*)d_ws;
    float* feat = ws;                                // N*128
    float* agg  = feat + (size_t)N * HC;             // N*128
    float* hbuf = agg + (size_t)N * HC;              // N*128
    float* ex   = hbuf + (size_t)N * HC;             // E*HEADS
    float* als  = ex + (size_t)E * HEADS;            // N*HEADS
    float* ald  = als + (size_t)N * HEADS;
    float* mbuf = ald + (size_t)N * HEADS;
    float* sbuf = mbuf + (size_t)N * HEADS;
    float* pooled = sbuf + (size_t)N * HEADS;        // G*HID
    float* cnt  = pooled + (size_t)GRP * HID;        // G

    const int BLK = 256;
    auto cdiv = [](long long a, long long b) { return (int)((a + b - 1) / b); };
    int gemm_blocks = cdiv(N / 16, 8);               // 8 waves (strips) per block

    auto run_edges = [&](const float* a_s, const float* a_d, const float* bias,
                         const float* g, const float* beta, int M, int H, float* out_h) {
        int C = M / H;
        score_kernel<<<cdiv((long long)N * H, BLK), BLK, 0, stream>>>(feat, a_s, a_d, als, ald, N, H, C);
        fill_kernel<<<cdiv((long long)N * H, BLK), BLK, 0, stream>>>(mbuf, -__builtin_huge_valf(), N * H);
        hipMemsetAsync(sbuf, 0, (size_t)N * H * sizeof(float), stream);
        hipMemsetAsync(agg, 0, (size_t)N * M * sizeof(float), stream);
        edge_max_kernel<<<cdiv((long long)E * H, BLK), BLK, 0, stream>>>(src, dst, als, ald, mbuf, E, H);
        edge_expsum_kernel<<<cdiv((long long)E * H, BLK), BLK, 0, stream>>>(src, dst, als, ald, mbuf, sbuf, ex, E, H);
        edge_agg_kernel<<<cdiv((long long)E * (M / 4), BLK), BLK, 0, stream>>>(src, dst, feat, ex, sbuf, agg, E, H, M);
        ln_elu_kernel<<<cdiv((long long)N * 32, BLK), BLK, 0, stream>>>(agg, bias, g, beta, out_h, N, M);
    };

    // Layer 0: 128 -> 4x32 concat
    gemm_wmma_lds<128, 8><<<gemm_blocks, 256, 0, stream>>>(x, W0, feat, N);
    run_edges(as0, ad0, b0, ln0g, ln0b, HC, HEADS, hbuf);
    // Layer 1: 128 -> 4x32 concat
    gemm_wmma_lds<128, 8><<<gemm_blocks, 256, 0, stream>>>(hbuf, W1, feat, N);
    run_edges(as1, ad1, b1, ln1g, ln1b, HC, HEADS, hbuf);
    // Layer 2: 128 -> 32, single head (mean over 1 head == identity)
    gemm_wmma_lds<128, 2><<<gemm_blocks, 256, 0, stream>>>(hbuf, W2, feat, N);
    run_edges(as2, ad2, b2, ln2g, ln2b, HID, 1, hbuf);

    // Global mean pool + classifier
    hipMemsetAsync(pooled, 0, (size_t)GRP * HID * sizeof(float), stream);
    hipMemsetAsync(cnt, 0, (size_t)GRP * sizeof(float), stream);
    pool_kernel<<<cdiv((long long)N * HID, BLK), BLK, 0, stream>>>(hbuf, batch, pooled, cnt, N);
    classify_kernel<<<1, GRP, 0, stream>>>(pooled, cnt, fc1w, fc1b, fc2w, fc2b, (float*)d_out);
}